// Layer_55697135894671
// MI455X (gfx1250) — compile-verified
//
#include <hip/hip_runtime.h>
#include <hip/hip_bf16.h>

typedef __attribute__((ext_vector_type(2))) float v2f;
typedef __attribute__((ext_vector_type(8))) float v8f;

// Fixed problem shape from the reference: B=1024, C=16, M=4096, N=8192.
#define B_DIM 1024
#define C_DIM 16
#define M_DIM 4096
#define N_DIM 8192
#define K_DIM 32
#define HALF_LOG_2PI 0.91893853320467274f

// ---------------------------------------------------------------------------
// Prep 1: U[b, 0..15] = x[b,c]^2 ; U[b, 16..31] = x[b,c]
// ---------------------------------------------------------------------------
__global__ void build_u_kernel(const float* __restrict__ x,
                               float* __restrict__ U) {
    int idx = blockIdx.x * blockDim.x + threadIdx.x;   // over B*C
    if (idx >= B_DIM * C_DIM) return;
    int b = idx >> 4;          // / 16
    int c = idx & 15;          // % 16
    float v = x[b * C_DIM + c];
    U[b * K_DIM + c]         = v * v;
    U[b * K_DIM + C_DIM + c] = v;
}

// ---------------------------------------------------------------------------
// Prep 2: gather mu/s via edges, build W (32 x N) and Koff (N).
//   W[c,n]    = -0.5 / s^2        (coefficient of x^2)
//   W[16+c,n] =  mu / s^2         (coefficient of x)
//   Koff[n]   = sum_c ( -0.5*mu^2/s^2 - log(s) - 0.5*log(2*pi) )
// ---------------------------------------------------------------------------
__global__ void build_w_kernel(const float* __restrict__ locs,
                               const float* __restrict__ scales,
                               const int*   __restrict__ edges,
                               float* __restrict__ W,
                               float* __restrict__ Koff) {
    int n = blockIdx.x * blockDim.x + threadIdx.x;
    if (n >= N_DIM) return;
    float ko = 0.0f;
#pragma unroll
    for (int c = 0; c < C_DIM; ++c) {
        int   e    = edges[c * N_DIM + n];
        float mu   = locs[c * M_DIM + e];
        float s    = scales[c * M_DIM + e] + 0.5f;
        float inv  = 1.0f / s;
        float inv2 = inv * inv;
        W[c * N_DIM + n]           = -0.5f * inv2;
        W[(C_DIM + c) * N_DIM + n] = mu * inv2;
        ko += -0.5f * mu * mu * inv2 - __logf(s) - HALF_LOG_2PI;
    }
    Koff[n] = ko;
}

// ---------------------------------------------------------------------------
// GEMM: out[b,n] = sum_k U[b,k]*W[k,n] + Koff[n]   (K = 32)
// One wave -> one 16x16 tile via 8x V_WMMA_F32_16X16X4_F32.
// Block = 8 waves = 128 rows (b) x 16 cols (n).
// Compile-time dims -> row strides become immediate IOFFSETs on loads/stores.
// ---------------------------------------------------------------------------
__global__ __launch_bounds__(256)
void wmma_gemm_kernel(const float* __restrict__ U,     // [B][32]
                      const float* __restrict__ W,     // [32][N]
                      const float* __restrict__ Koff,  // [N]
                      float* __restrict__ out) {       // [B][N]
    const int lane = threadIdx.x & 31;
    const int wave = threadIdx.x >> 5;

    constexpr int TILES_N = N_DIM / 16;        // 512
    const int n_tile  = blockIdx.x % TILES_N;
    const int b_group = blockIdx.x / TILES_N;

    const int b0 = (b_group * 8 + wave) * 16;
    const int n0 = n_tile * 16;

    const int half = lane >> 4;   // 0: lanes 0-15, 1: lanes 16-31
    const int li   = lane & 15;

    // Per-lane base pointers; all K-steps become immediate offsets.
    const float* Ua = U + (b0 + li) * K_DIM + half * 2;
    const float* Wb = W + half * 2 * N_DIM + (n0 + li);

    v8f acc = {};
#pragma unroll
    for (int k0 = 0; k0 < K_DIM; k0 += 4) {
        // A fragment: 16x4 tile of U (rows = b, K = k0+half*2 .. +1)
        v2f a;
        a.x = Ua[k0];
        a.y = Ua[k0 + 1];
        // B fragment: 4x16 tile of W (K rows, cols = n)
        v2f bf;
        bf.x = Wb[k0 * N_DIM];
        bf.y = Wb[(k0 + 1) * N_DIM];
        acc = __builtin_amdgcn_wmma_f32_16x16x4_f32(
            /*neg_a=*/false, a, /*neg_b=*/false, bf,
            /*c_mod=*/(short)0, acc, /*reuse_a=*/false, /*reuse_b=*/false);
    }

    const float ko = Koff[n0 + li];
    // C/D layout: VGPR i -> row b0+i (lanes 0-15) / b0+i+8 (lanes 16-31).
    float* o = out + (b0 + half * 8) * N_DIM + (n0 + li);
#pragma unroll
    for (int i = 0; i < 8; ++i) {
        o[i * N_DIM] = acc[i] + ko;   // i*N_DIM*4 fits 24-bit IOFFSET
    }
}

// ---------------------------------------------------------------------------
extern "C" void kernel_launch(void* const* d_in, const int* in_sizes, int n_in,
                              void* d_out, int out_size, void* d_ws, size_t ws_size,
                              hipStream_t stream) {
    const float* x      = (const float*)d_in[0];  // (B, C)
    const float* locs   = (const float*)d_in[1];  // (C, M)
    const float* scales = (const float*)d_in[2];  // (C, M)
    const int*   edges  = (const int*)d_in[3];    // (C, N)
    float*       out    = (float*)d_out;          // (B, N)

    // Workspace layout: U [B*32] | W [32*N] | Koff [N]   (~1.2 MB)
    float* U    = (float*)d_ws;
    float* W    = U + (size_t)B_DIM * K_DIM;
    float* Koff = W + (size_t)K_DIM * N_DIM;

    build_u_kernel<<<(B_DIM * C_DIM + 255) / 256, 256, 0, stream>>>(x, U);
    build_w_kernel<<<(N_DIM + 255) / 256, 256, 0, stream>>>(locs, scales, edges,
                                                            W, Koff);

    // 8 waves/block -> 128 b-rows x 16 n-cols per block
    constexpr int GRID = (N_DIM / 16) * (B_DIM / 128);   // 4096
    wmma_gemm_kernel<<<GRID, 256, 0, stream>>>(U, W, Koff, out);
}